// GraphAttentionWithAdjacency_27608049779020
// MI455X (gfx1250) — compile-verified
//
#include <hip/hip_runtime.h>
#include <hip/hip_bf16.h>

// CDNA5 / gfx1250, wave32. fp32 WMMA (V_WMMA_F32_16X16X4_F32) for the main GEMM.

typedef __attribute__((ext_vector_type(2))) float v2f;
typedef __attribute__((ext_vector_type(8))) float v8f;

#define SEQ_LEN 2048
#define BATCH   8
#define FEAT    128      // IN_FEAT == UNITS == 128
#define ROWS    (BATCH * SEQ_LEN)   // 16384

// -----------------------------------------------------------------------------
// Kernel 1: nf = x @ W   (M=16384, K=128, N=128), fp32 WMMA 16x16x4.
// Block = 256 threads = 8 waves. Wave w computes the 16x16 tile at
// (m_base = blockIdx.x*16, n_base = w*16), looping K in steps of 4.
// Fragment layouts per CDNA5 ISA 7.12.2 (wave32):
//   A 16x4 : lane l, reg j -> A[m_base + (l&15)][kk + 2*(l>>4) + j]
//   B 4x16 : lane l, reg j -> W[kk + j + 2*(l>>4)][n_base + (l&15)]
//   D 16x16: lane l, reg r -> D[m_base + r + 8*(l>>4)][n_base + (l&15)]
// -----------------------------------------------------------------------------
__global__ __launch_bounds__(256) void gemm_nf_wmma(const float* __restrict__ x,
                                                    const float* __restrict__ W,
                                                    float* __restrict__ nf) {
  const int lane = threadIdx.x & 31;
  const int wave = threadIdx.x >> 5;
  const int half = lane >> 4;   // 0 or 1
  const int lm   = lane & 15;
  const int m_base = blockIdx.x * 16;
  const int n_base = wave * 16;

  const float* arow = x + (size_t)(m_base + lm) * FEAT + 2 * half;  // 8B aligned
  const float* bcol = W + (size_t)(2 * half) * FEAT + n_base + lm;

  v8f c = {};
#pragma unroll 4
  for (int kk = 0; kk < FEAT; kk += 4) {
    const float2 a2 = *(const float2*)(arow + kk);        // K = kk+2*half {+0,+1}
    v2f a; a[0] = a2.x; a[1] = a2.y;
    v2f b;
    b[0] = bcol[(size_t)(kk + 0) * FEAT];                 // K = kk + 2*half + 0
    b[1] = bcol[(size_t)(kk + 1) * FEAT];                 // K = kk + 2*half + 1
    c = __builtin_amdgcn_wmma_f32_16x16x4_f32(
        /*neg_a=*/false, a, /*neg_b=*/false, b,
        /*c_mod=*/(short)0, c, /*reuse_a=*/false, /*reuse_b=*/false);
  }

  float* drow = nf + (size_t)(m_base + 8 * half) * FEAT + n_base + lm;
#pragma unroll
  for (int r = 0; r < 8; ++r) drow[(size_t)r * FEAT] = c[r];
}

// -----------------------------------------------------------------------------
// Kernel 2: e_src[g] = nf[g,:] . a_src ; e_dst[g] = nf[g,:] . a_dst
// One wave32 per row; lane covers 4 features; xor-shuffle reduction.
// -----------------------------------------------------------------------------
__global__ __launch_bounds__(256) void row_dots(const float* __restrict__ nf,
                                                const float* __restrict__ attn_k,
                                                float* __restrict__ e_src,
                                                float* __restrict__ e_dst) {
  const int lane = threadIdx.x & 31;
  const int wave = threadIdx.x >> 5;
  const int row  = blockIdx.x * 8 + wave;

  const float4 v  = *(const float4*)(nf + (size_t)row * FEAT + lane * 4);
  const float4 as = *(const float4*)(attn_k + lane * 4);           // a_src
  const float4 ad = *(const float4*)(attn_k + FEAT + lane * 4);    // a_dst

  float ps = v.x * as.x + v.y * as.y + v.z * as.z + v.w * as.w;
  float pd = v.x * ad.x + v.y * ad.y + v.z * ad.z + v.w * ad.w;
#pragma unroll
  for (int off = 16; off > 0; off >>= 1) {
    ps += __shfl_xor(ps, off, 32);
    pd += __shfl_xor(pd, off, 32);
  }
  if (lane == 0) { e_src[row] = ps; e_dst[row] = pd; }
}

// -----------------------------------------------------------------------------
// Kernel 3: banded softmax + attn @ nf.
// Adjacency is a 5-diagonal band: weight 1.0 for |d|<=1, 0.5 for |d|==2,
// -FLT_MAX elsewhere -> softmax over at most 5 entries per row.
// One wave32 per output row; each lane owns 4 of the 128 features.
// -----------------------------------------------------------------------------
__global__ __launch_bounds__(256) void band_attn(const float* __restrict__ nf,
                                                 const float* __restrict__ e_src,
                                                 const float* __restrict__ e_dst,
                                                 float* __restrict__ out) {
  const int lane = threadIdx.x & 31;
  const int wave = threadIdx.x >> 5;
  const int g = blockIdx.x * 8 + wave;   // global row, 0..16383
  const int b = g >> 11;                 // / SEQ_LEN
  const int i = g & (SEQ_LEN - 1);

  const float NEG = -3.402823466e+38f;   // finfo(f32).min
  const float esi = e_src[g];

  float sc[5];
  float m = NEG;
#pragma unroll
  for (int d = 0; d < 5; ++d) {
    const int j = i + d - 2;
    float s = NEG;
    if (j >= 0 && j < SEQ_LEN) {
      float t = esi + e_dst[(b << 11) + j];
      t = (t > 0.0f) ? t : 0.2f * t;                 // leaky_relu(0.2)
      const float w = (d == 0 || d == 4) ? 0.5f : 1.0f;  // adj weight
      s = t * w;
    }
    sc[d] = s;
    m = fmaxf(m, s);
  }

  float p[5];
  float denom = 0.0f;
#pragma unroll
  for (int d = 0; d < 5; ++d) {
    const float e = (sc[d] <= -3.0e38f) ? 0.0f : __expf(sc[d] - m);
    p[d] = e;
    denom += e;
  }
  const float inv = 1.0f / denom;

  float4 acc = {0.0f, 0.0f, 0.0f, 0.0f};
#pragma unroll
  for (int d = 0; d < 5; ++d) {
    const int j = i + d - 2;
    if (j >= 0 && j < SEQ_LEN && p[d] != 0.0f) {
      const float a = p[d] * inv;
      const float4 v = *(const float4*)(nf + (size_t)((b << 11) + j) * FEAT + lane * 4);
      acc.x += a * v.x; acc.y += a * v.y; acc.z += a * v.z; acc.w += a * v.w;
    }
  }
  *(float4*)(out + (size_t)g * FEAT + lane * 4) = acc;
}

// -----------------------------------------------------------------------------
extern "C" void kernel_launch(void* const* d_in, const int* in_sizes, int n_in,
                              void* d_out, int out_size, void* d_ws, size_t ws_size,
                              hipStream_t stream) {
  const float* x      = (const float*)d_in[0];   // (8, 2048, 128)
  const float* W      = (const float*)d_in[1];   // (128, 128)
  const float* attn_k = (const float*)d_in[2];   // (256, 1)

  float* nf    = (float*)d_ws;                   // 16384 x 128 f32 = 8 MB
  float* e_src = nf + (size_t)ROWS * FEAT;       // 16384 f32
  float* e_dst = e_src + ROWS;                   // 16384 f32

  // 1) nf = x @ W  (WMMA fp32): 1024 blocks x (8 waves x 16x16 tiles)
  gemm_nf_wmma<<<ROWS / 16, 256, 0, stream>>>(x, W, nf);
  // 2) row dot products (wave per row)
  row_dots<<<ROWS / 8, 256, 0, stream>>>(nf, attn_k, e_src, e_dst);
  // 3) banded softmax + weighted sum (wave per row)
  band_attn<<<ROWS / 8, 256, 0, stream>>>(nf, e_src, e_dst, (float*)d_out);
}